// RGCN_68049461838043
// MI455X (gfx1250) — compile-verified
//
#include <hip/hip_runtime.h>
#include <hip/hip_bf16.h>
#include <math.h>

// RGCN on MI455X (gfx1250, wave32).
// Strategy: aggregate-then-project so the per-edge work touches only an
// L2-resident [N,128] f32 buffer (25.6MB << 192MB L2), looped over the 8
// relations. GEMMs use V_WMMA_F32_16X16X4_F32 with W staged in LDS,
// pre-swizzled into the exact WMMA fragment layout.

typedef __attribute__((ext_vector_type(2))) float v2f;
typedef __attribute__((ext_vector_type(8))) float v8f;

#define DIM   128
#define NRELS 8

// ---------------------------------------------------------------- gather x = emb[h]
__global__ void rgcn_gather(const float* __restrict__ emb, const int* __restrict__ h,
                            float* __restrict__ x, int N) {
  int n = blockIdx.x;
  if (n < N)
    x[(size_t)n * DIM + threadIdx.x] = emb[(size_t)h[n] * DIM + threadIdx.x];
}

// --------------------------------------------- scatter: Ab[dst] += norm * x[src] for rel==r
// blockDim = 128 (one thread per channel); one edge per block iteration.
// rel/src/dst/norm are uniform per iteration -> scalar loads; x row read is a
// coalesced 512B load; atomic add is native global_atomic_add_f32 hitting L2.
__global__ void rgcn_scatter(const int* __restrict__ src, const int* __restrict__ dst,
                             const int* __restrict__ rel, const float* __restrict__ norm,
                             const float* __restrict__ x, float* __restrict__ Ab,
                             int r, int E) {
  int c = threadIdx.x;
  for (int e = blockIdx.x; e < E; e += gridDim.x) {
    if (rel[e] != r) continue;
    float v = norm[e] * x[(size_t)src[e] * DIM + c];
    __hip_atomic_fetch_add(&Ab[(size_t)dst[e] * DIM + c], v,
                           __ATOMIC_RELAXED, __HIP_MEMORY_SCOPE_AGENT);
  }
}

// ---------------------------------------------------------------- WMMA f32 GEMM
// Cout[M,128] = act( A[M,128] @ B[128,128] + (flags&1 ? bias : Cin) )
// flags bit0: init accumulators from broadcast bias (self-loop GEMM)
//       bit1: apply tanh on store
// 256 threads = 8 waves; wave w handles rows [ (8*blk+w)*16 , +16 ), all 128 cols
// => 8 accumulator tiles of 16x16 (v8f each), K marched in steps of 4.
//
// Fragment layouts (ISA 7.12.2, 32-bit, wave32):
//  A 16x4 : lane l<16 -> row l, {K0,K0+1}; lane l>=16 -> row l-16, {K0+2,K0+3}
//  B 4x16 : lane l<16 -> col l, {K0,K0+1}; lane l>=16 -> col l-16, {K0+2,K0+3}
//  C/D    : VGPR v, lanes 0-15 -> row v, lanes 16-31 -> row v+8; col = lane&15
__global__ __launch_bounds__(256)
void rgcn_gemm(const float* __restrict__ A, const float* __restrict__ B,
               const float* __restrict__ Cin, const float* __restrict__ bias,
               float* __restrict__ Cout, int M, int flags) {
  // Bs[t][s][lane] : B fragment (v2f) for N-tile t (cols 16t..16t+15), k-step s (K=4s..4s+3)
  __shared__ v2f Bs[8 * 32 * 32];  // 64 KB

  for (int i = threadIdx.x; i < 8 * 32 * 32; i += 256) {
    int lane = i & 31;
    int s    = (i >> 5) & 31;
    int t    = i >> 10;
    int kB   = 4 * s + ((lane & 16) ? 2 : 0);
    int n    = 16 * t + (lane & 15);
    v2f b;
    b.x = B[(size_t)kB * DIM + n];
    b.y = B[(size_t)(kB + 1) * DIM + n];
    Bs[i] = b;
  }
  __syncthreads();

  int wave = threadIdx.x >> 5;
  int lane = threadIdx.x & 31;
  int m0   = (blockIdx.x * 8 + wave) * 16;
  if (m0 >= M) return;  // wave-uniform: EXEC stays all-1s for WMMA

  int ncol  = lane & 15;
  int khalf = (lane & 16) ? 2 : 0;
  int rbase = m0 + ((lane & 16) ? 8 : 0);  // C/D row base for this half-wave

  v8f acc[8];
  if (flags & 1) {
#pragma unroll
    for (int t = 0; t < 8; ++t) {
      float bv = bias[16 * t + ncol];
#pragma unroll
      for (int v = 0; v < 8; ++v) acc[t][v] = bv;
    }
  } else {
#pragma unroll
    for (int t = 0; t < 8; ++t)
#pragma unroll
      for (int v = 0; v < 8; ++v)
        acc[t][v] = Cin[(size_t)(rbase + v) * DIM + 16 * t + ncol];
  }

  const float* Arow = A + (size_t)(m0 + ncol) * DIM;  // lanes 0-15 & 16-31 share rows
  for (int s = 0; s < 32; ++s) {
    v2f a = *(const v2f*)(Arow + 4 * s + khalf);  // 8B contiguous -> global_load_b64
#pragma unroll
    for (int t = 0; t < 8; ++t) {
      v2f b = Bs[t * 1024 + s * 32 + lane];       // ds_load_b64
      acc[t] = __builtin_amdgcn_wmma_f32_16x16x4_f32(
          false, a, false, b, (short)0, acc[t], false, false);
    }
  }

#pragma unroll
  for (int t = 0; t < 8; ++t)
#pragma unroll
    for (int v = 0; v < 8; ++v) {
      float r = acc[t][v];
      if (flags & 2) r = tanhf(r);
      Cout[(size_t)(rbase + v) * DIM + 16 * t + ncol] = r;
    }
}

// ---------------------------------------------------------------- host driver
extern "C" void kernel_launch(void* const* d_in, const int* in_sizes, int n_in,
                              void* d_out, int out_size, void* d_ws, size_t ws_size,
                              hipStream_t stream) {
  const int*   h     = (const int*)d_in[0];
  const int*   src   = (const int*)d_in[1];
  const int*   dst   = (const int*)d_in[2];
  const int*   rel   = (const int*)d_in[3];
  const float* norm  = (const float*)d_in[4];
  const float* emb   = (const float*)d_in[5];
  const float* W0    = (const float*)d_in[6];
  const float* loop0 = (const float*)d_in[7];
  const float* b0    = (const float*)d_in[8];
  const float* W1    = (const float*)d_in[9];
  const float* loop1 = (const float*)d_in[10];
  const float* b1    = (const float*)d_in[11];
  float*       out   = (float*)d_out;

  const int N = in_sizes[0];
  const int E = in_sizes[1];
  const size_t matBytes = (size_t)N * DIM * sizeof(float);

  char*  ws  = (char*)d_ws;
  float* x   = (float*)(ws);                 // layer-0 input  (emb[h])
  float* y   = (float*)(ws + matBytes);      // layer-0 output / layer-1 input
  float* Ab  = (float*)(ws + 2 * matBytes);  // per-relation scatter accumulator
  float* agg = out;                          // running accumulator lives in d_out

  const int gemmBlocks = (N + 8 * 16 - 1) / (8 * 16);

  rgcn_gather<<<N, DIM, 0, stream>>>(emb, h, x, N);

  for (int layer = 0; layer < 2; ++layer) {
    const float* xin  = layer ? y : x;
    const float* W    = layer ? W1 : W0;
    const float* lw   = layer ? loop1 : loop0;
    const float* bias = layer ? b1 : b0;
    float*       yout = layer ? out : y;
    const int    actF = layer ? 0 : 2;  // tanh only after layer 0

    // self-loop: agg = xin @ lw + bias   (init accumulators from bias)
    rgcn_gemm<<<gemmBlocks, 256, 0, stream>>>(xin, lw, agg, bias, agg, N, /*flags=*/1);

    for (int r = 0; r < NRELS; ++r) {
      hipMemsetAsync(Ab, 0, matBytes, stream);
      rgcn_scatter<<<4096, DIM, 0, stream>>>(src, dst, rel, norm, xin, Ab, r, E);
      const bool last = (r == NRELS - 1);
      rgcn_gemm<<<gemmBlocks, 256, 0, stream>>>(
          Ab, W + (size_t)r * DIM * DIM, agg, bias,
          last ? yout : agg, N, last ? actF : 0);
    }
  }
}